// StreamingDurationProjector_63788854280284
// MI455X (gfx1250) — compile-verified
//
#include <hip/hip_runtime.h>
#include <stdint.h>

// StreamingDurationProjector for MI455X (gfx1250, wave32).
// One lane per row, 64 single-wave blocks. Input tiles are staged into LDS
// with CDNA5 async copies (global_load_async_to_lds_b128, ASYNCcnt-tracked,
// double-buffered, issued one tile ahead) so the latency-bound serial scan
// never waits on global memory and all global loads are coalesced.

namespace {

constexpr int   kB         = 2048;
constexpr int   kU         = 4096;
constexpr float kBudgetPos = 24.0f;
constexpr float kBudgetNeg = 24.0f;

constexpr int kRowsPerBlk = 32;                    // lanes == rows per block
constexpr int kTileCols   = 32;                    // 128 B per row per tile
constexpr int kTiles      = kU / kTileCols;        // 128
constexpr int kStrideDw   = 36;                    // LDS row stride (dwords): 144 B,
                                                   // 16B-aligned, max 2-way bank conflict
constexpr int kBufDw      = kRowsPerBlk * kStrideDw;

typedef float float4v __attribute__((ext_vector_type(4)));
typedef int   int4v   __attribute__((ext_vector_type(4)));

// LDS byte address (wave-relative) of a __shared__ object, for the async VDST operand.
__device__ __forceinline__ uint32_t lds_addr(const void* p) {
    return (uint32_t)(uintptr_t)(__attribute__((address_space(3))) const void*)p;
}

// CDNA5 async copy: 16 B per lane, global -> LDS, tracked by ASYNCcnt.
__device__ __forceinline__ void async_load_b128(uint32_t lds_byte, uint64_t gaddr) {
    asm volatile("global_load_async_to_lds_b128 %0, %1, off"
                 :: "v"(lds_byte), "v"(gaddr)
                 : "memory");
}

__device__ __forceinline__ void wait_asynccnt_le32() {
    asm volatile("s_wait_asynccnt 32" ::: "memory");
}
__device__ __forceinline__ void wait_asynccnt_0() {
    asm volatile("s_wait_asynccnt 0" ::: "memory");
}
__device__ __forceinline__ void wait_dscnt_0() {
    asm volatile("s_wait_dscnt 0" ::: "memory");
}

// One step of the reference recurrence. Mirrors _project_row.step exactly.
__device__ __forceinline__ void scan_step(float d, int s, float m, int sm_i,
                                          float& c, float& off,
                                          float& proj_out, float& pcm_out,
                                          float& mat_out)
{
    float src_count = fmaxf(0.0f, rintf((float)s));   // round-half-even; exact for ints
    float total     = fmaxf(0.0f, d + c);
    float frames0   = fmaxf(1.0f, floorf(total));
    float anchor    = fmaxf(1.0f, src_count);
    float lower     = fmaxf(1.0f, ceilf(anchor - (kBudgetNeg + off)));
    float upper     = fmaxf(lower, floorf(anchor + (kBudgetPos - off)));
    float frames    = fminf(fmaxf(frames0, lower), upper);   // clip

    bool is_commit  = m > 0.5f;          // float mask semantics
    bool is_speech  = sm_i > 0;          // float(sm) > 0.5 for integer sm
    bool active     = is_commit && is_speech;

    float proj = is_commit ? (is_speech ? frames : src_count) : 0.0f;
    float pcm  = proj * m;               // projected * commit_mask

    proj_out = proj;
    pcm_out  = pcm;
    // materialized = dur + stop_grad(pcm - dur): keep the exact fp expression.
    mat_out  = d + (pcm - d);

    c   = active ? (total - frames)        : c;
    off = active ? (off + frames - anchor) : off;
}

__global__ __launch_bounds__(32)
void sdp_scan_kernel(const float* __restrict__ dur,
                     const int*   __restrict__ src,
                     const float* __restrict__ cm,
                     const int*   __restrict__ scm,
                     const float* __restrict__ res_prev,
                     const float* __restrict__ off_prev,
                     float*       __restrict__ out)
{
    __shared__ __align__(16) float s_dur[2][kBufDw];
    __shared__ __align__(16) int   s_src[2][kBufDw];
    __shared__ __align__(16) float s_cm [2][kBufDw];
    __shared__ __align__(16) int   s_scm[2][kBufDw];

    const int lane = threadIdx.x;
    const int row0 = blockIdx.x * kRowsPerBlk;
    const int row  = row0 + lane;

    // ---- async-issue geometry: each instruction moves 4 rows x 128 B, coalesced ----
    const int lr = lane >> 3;            // which of 4 rows this lane serves
    const int lc = lane & 7;             // which 16B chunk within the row's 128B
    const uint64_t g_lane0 = ((uint64_t)(row0 + lr) * kU + (uint32_t)lc * 4u) * 4u;
    const uint32_t l_lane0 = ((uint32_t)lr * kStrideDw + (uint32_t)lc * 4u) * 4u;

    const uint64_t gdur = (uint64_t)(uintptr_t)dur + g_lane0;
    const uint64_t gsrc = (uint64_t)(uintptr_t)src + g_lane0;
    const uint64_t gcm  = (uint64_t)(uintptr_t)cm  + g_lane0;
    const uint64_t gscm = (uint64_t)(uintptr_t)scm + g_lane0;

    const uint32_t ldur = lds_addr(&s_dur[0][0]) + l_lane0;
    const uint32_t lsrc = lds_addr(&s_src[0][0]) + l_lane0;
    const uint32_t lcm  = lds_addr(&s_cm [0][0]) + l_lane0;
    const uint32_t lscm = lds_addr(&s_scm[0][0]) + l_lane0;

    auto issue_tile = [&](int t, int buf) {
        const uint64_t gtile = (uint64_t)t * (kTileCols * 4);       // +128 B per tile
        const uint32_t lbuf  = (uint32_t)buf * (uint32_t)(kBufDw * 4);
        #pragma unroll
        for (int j = 0; j < 8; ++j) {                               // 4 rows per issue
            const uint64_t gj = gtile + (uint64_t)j * (4ull * kU * 4ull);
            const uint32_t lj = lbuf + (uint32_t)j * (4u * kStrideDw * 4u);
            async_load_b128(ldur + lj, gdur + gj);
            async_load_b128(lsrc + lj, gsrc + gj);
            async_load_b128(lcm  + lj, gcm  + gj);
            async_load_b128(lscm + lj, gscm + gj);
        }
    };

    const long base = (long)row * kU;
    const long BU   = (long)kB * kU;
    float4v* __restrict__ matv = (float4v*)(out            + base);
    float4v* __restrict__ prjv = (float4v*)(out + BU       + base);
    float4v* __restrict__ cchv = (float4v*)(out + 2 * BU   + base);

    float c   = res_prev[row];
    float off = rintf(off_prev[row]);    // off0 = jnp.round(prefix_offset_prev)

    issue_tile(0, 0);

    for (int t = 0; t < kTiles; ++t) {
        const int buf = t & 1;
        if (t + 1 < kTiles) {
            wait_dscnt_0();              // buffer (t+1)&1 fully consumed (WAR safety)
            issue_tile(t + 1, (t + 1) & 1);
            wait_asynccnt_le32();        // in-order completion => tile t resident
        } else {
            wait_asynccnt_0();
        }

        const float* drow = &s_dur[buf][lane * kStrideDw];
        const int*   srow = &s_src[buf][lane * kStrideDw];
        const float* mrow = &s_cm [buf][lane * kStrideDw];
        const int*   trow = &s_scm[buf][lane * kStrideDw];

        #pragma unroll
        for (int q = 0; q < 8; ++q) {
            float4v d4 = *(const float4v*)(drow + q * 4);   // ds_load_b128
            int4v   s4 = *(const int4v*)  (srow + q * 4);
            float4v m4 = *(const float4v*)(mrow + q * 4);
            int4v   t4 = *(const int4v*)  (trow + q * 4);

            float4v proj4, pcm4, mat4;
            #pragma unroll
            for (int k = 0; k < 4; ++k) {
                float pj, pc, mt;
                scan_step(d4[k], s4[k], m4[k], t4[k], c, off, pj, pc, mt);
                proj4[k] = pj;
                pcm4[k]  = pc;
                mat4[k]  = mt;
            }

            const int vi = t * 8 + q;
            matv[vi] = mat4;                                // global_store_b128 x3
            prjv[vi] = proj4;
            cchv[vi] = pcm4;
        }
    }

    // residual_next[:, None] and offset_next[:, None], flat-concatenated.
    out[3 * BU + row]      = c;
    out[3 * BU + kB + row] = off;
}

} // namespace

extern "C" void kernel_launch(void* const* d_in, const int* in_sizes, int n_in,
                              void* d_out, int out_size, void* d_ws, size_t ws_size,
                              hipStream_t stream)
{
    (void)in_sizes; (void)n_in; (void)d_ws; (void)ws_size; (void)out_size;

    const float* dur      = (const float*)d_in[0];  // unit_duration_exec  (B,U) f32
    const int*   src      = (const int*)  d_in[1];  // source_duration_obs (B,U) i32
    const float* cm       = (const float*)d_in[2];  // unit_mask           (B,U) f32
    const int*   scm      = (const int*)  d_in[3];  // speech_commit_mask  (B,U) i32
    const float* res_prev = (const float*)d_in[4];  // residual_prev       (B,1) f32
    const float* off_prev = (const float*)d_in[5];  // prefix_offset_prev  (B,1) f32

    dim3 grid(kB / kRowsPerBlk);   // 64 single-wave workgroups -> 64 WGPs
    dim3 block(kRowsPerBlk);       // one wave32; lane == row
    hipLaunchKernelGGL(sdp_scan_kernel, grid, block, 0, stream,
                       dur, src, cm, scm, res_prev, off_prev, (float*)d_out);
}